// OneLayerRNN_13288628814243
// MI455X (gfx1250) — compile-verified
//
#include <hip/hip_runtime.h>
#include <math.h>
#include <stdint.h>

typedef __attribute__((ext_vector_type(2))) float v2f;
typedef __attribute__((ext_vector_type(8))) float v8f;
typedef __attribute__((ext_vector_type(4))) unsigned u32x4;
typedef __attribute__((ext_vector_type(8))) int i32x8;
typedef __attribute__((ext_vector_type(4))) int i32x4;

namespace {
constexpr int kB = 64;
constexpr int kS = 512;
constexpr int kI = 1024;
constexpr int kH = 1024;
constexpr int kM1 = kB * kS;       // 32768 rows in the input projection
constexpr int kScanBlocks = 64;    // persistent blocks, one 16-col tile each
constexpr int kLdsStride = 1032;   // 1024 + 2 dwords pad per 256 dwords
constexpr int kLdsFloats = 16 * kLdsStride;  // 16512 floats = 66048 B
}

// D = A(16x4,f32) * B(4x16,f32) + C(16x16,f32)
__device__ __forceinline__ v8f wmma_f32_k4(v2f a, v2f b, v8f c) {
  return __builtin_amdgcn_wmma_f32_16x16x4_f32(
      /*neg_a=*/false, a, /*neg_b=*/false, b,
      /*c_mod=*/(short)0, c, /*reuse_a=*/false, /*reuse_b=*/false);
}

// ---------------------------------------------------------------------------
// Kernel 1: xw[m,n] = sum_k x[m,k] * Wih[n,k] + bih[n], written into d_out.
// One wave computes a 16(M) x 64(N) strip: A fragment reused across 4 tiles.
// ---------------------------------------------------------------------------
__global__ __launch_bounds__(256) void rnn_xproj_wmma(
    const float* __restrict__ x,    // [kM1, kI]
    const float* __restrict__ Wih,  // [kH, kI]
    const float* __restrict__ bih,  // [kH]
    float* __restrict__ xw)         // [kM1, kH]
{
  const int lane = threadIdx.x & 31;
  const int wave = threadIdx.x >> 5;
  const int gw = blockIdx.x * 8 + wave;  // 2048 waves total
  const int mTile = gw >> 4;             // 128 M-tiles of 16 rows
  const int nStrip = gw & 15;            // 16 N-strips of 64 cols
  const int m0 = mTile * 16;
  const int n0 = nStrip * 64;

  const int mA = lane & 15;
  const int kk = (lane >> 4) * 2;
  const float* aRow = x + (size_t)(m0 + mA) * kI + kk;
  const float* bCol = Wih + (size_t)(n0 + (lane & 15)) * kI + kk;

  v8f c0 = {}, c1 = {}, c2 = {}, c3 = {};
  for (int k0 = 0; k0 < kI; k0 += 4) {
    v2f a = *(const v2f*)(aRow + k0);
    v2f b0 = *(const v2f*)(bCol + (size_t)0 * kI + k0);
    v2f b1 = *(const v2f*)(bCol + (size_t)16 * kI + k0);
    v2f b2 = *(const v2f*)(bCol + (size_t)32 * kI + k0);
    v2f b3 = *(const v2f*)(bCol + (size_t)48 * kI + k0);
    c0 = wmma_f32_k4(a, b0, c0);
    c1 = wmma_f32_k4(a, b1, c1);
    c2 = wmma_f32_k4(a, b2, c2);
    c3 = wmma_f32_k4(a, b3, c3);
  }

  const int rowBase = m0 + (lane >> 4) * 8;
  const int col = lane & 15;
#pragma unroll
  for (int t = 0; t < 4; ++t) {
    v8f c = (t == 0) ? c0 : (t == 1) ? c1 : (t == 2) ? c2 : c3;
    const int n = n0 + t * 16 + col;
    const float bias = bih[n];
#pragma unroll
    for (int i = 0; i < 8; ++i) {
      xw[(size_t)(rowBase + i) * kH + n] = c[i] + bias;
    }
  }
}

// ---------------------------------------------------------------------------
// Kernel 2: persistent scan. 64 blocks x 128 threads; block b owns output
// columns [16b, 16b+16). Its W_hh slice (16 rows x 1024 K) is TDM-loaded into
// LDS once (padded layout for bank spread), then all 512 timesteps run inside
// the kernel with a device-wide barrier between steps.
// ---------------------------------------------------------------------------
__global__ __launch_bounds__(128) void rnn_scan_persistent(
    const float* __restrict__ h0,   // [kB, kH]
    const float* __restrict__ Whh,  // [kH, kH]
    const float* __restrict__ bhh,  // [kH]
    float* __restrict__ out,        // [kB, kS, kH], xw pre-stored, tanh in place
    float* __restrict__ hb0,        // [kB, kH] ping
    float* __restrict__ hb1,        // [kB, kH] pong
    unsigned* __restrict__ bar)     // device-wide barrier counter (zeroed)
{
  extern __shared__ float bsh[];  // [16][kLdsStride] padded W_hh tile
  const int lane = threadIdx.x & 31;
  const int wave = threadIdx.x >> 5;
  const int n0 = blockIdx.x * 16;  // this block's output columns
  const int m0 = wave * 16;        // this wave's batch rows

  // ---- one-time stage of W_hh[n0:n0+16, :] into LDS via the TDM ----
#if __has_builtin(__builtin_amdgcn_tensor_load_to_lds)
  if (threadIdx.x < 32) {  // one wave issues the TDM op (EXEC ignored by TDM)
    const unsigned lds_base = (unsigned)(uintptr_t)(&bsh[0]);
    const unsigned long long ga =
        (unsigned long long)(uintptr_t)(Whh + (size_t)n0 * kH);
    u32x4 g0;
    g0.x = 1u;                                   // count=1 (valid), user mode
    g0.y = lds_base;                             // lds_addr
    g0.z = (unsigned)(ga & 0xffffffffu);         // global_addr[31:0]
    g0.w = (unsigned)((ga >> 32) & 0x01ffffffu)  // global_addr[56:32]
           | (2u << 30);                         // descriptor type = 2
    i32x8 g1;
    g1[0] = (int)((2u << 16)     // data_size = 4 bytes
                | (1u << 20)     // pad_enable
                | (7u << 22)     // pad_interval = 256 dwords
                | (1u << 25));   // pad_amount  = 2 dwords
    g1[1] = (int)(1024u << 16);  // tensor_dim0 = 1024 (bits 79:48)
    g1[2] = (int)(16u << 16);    // tensor_dim1 = 16   (bits 111:80)
    g1[3] = (int)(1024u << 16);  // tile_dim0   = 1024 (bits 127:112)
    g1[4] = 16;                  // tile_dim1 = 16, tile_dim2 = 0
    g1[5] = 1024;                // tensor_dim0_stride (lo 32)
    g1[6] = 0;                   // stride hi, dim1_stride lo
    g1[7] = 0;
    const i32x4 gz4 = {0, 0, 0, 0};
    const i32x8 gz8 = {0, 0, 0, 0, 0, 0, 0, 0};
    __builtin_amdgcn_tensor_load_to_lds(g0, g1, gz4, gz4, gz8, 0);
    __builtin_amdgcn_s_wait_tensorcnt(0);
  }
#else
  for (int idx = threadIdx.x; idx < 16 * 1024; idx += 128) {
    const int n = idx >> 10;
    const int k = idx & 1023;
    bsh[n * kLdsStride + k + 2 * (k >> 8)] = Whh[(size_t)(n0 + n) * kH + k];
  }
#endif
  __syncthreads();

  const int mA = lane & 15;
  const int kk = (lane >> 4) * 2;
  const int bRow = (lane & 15) * kLdsStride + kk;  // B fragment base (words)
  const int nCol = n0 + (lane & 15);
  const float bias = bhh[nCol];
  const int rowBase = m0 + (lane >> 4) * 8;

  for (int s = 0; s < kS; ++s) {
    const float* hin = (s == 0) ? h0 : ((s & 1) ? hb0 : hb1);
    float* hout = (s & 1) ? hb1 : hb0;
    const float* aRow = hin + (size_t)(m0 + mA) * kH + kk;

    // 4 independent accumulator chains over K for WMMA pipelining.
    v8f c0 = {}, c1 = {}, c2 = {}, c3 = {};
    for (int k0 = 0; k0 < kH; k0 += 16) {
      const int ka = bRow + k0 + 2 * (k0 >> 8);  // padded-LDS adjusted index
      v2f a0 = *(const v2f*)(aRow + k0);
      v2f a1 = *(const v2f*)(aRow + k0 + 4);
      v2f a2 = *(const v2f*)(aRow + k0 + 8);
      v2f a3 = *(const v2f*)(aRow + k0 + 12);
      v2f b0 = *(const v2f*)(bsh + ka);
      v2f b1 = *(const v2f*)(bsh + ka + 4);
      v2f b2 = *(const v2f*)(bsh + ka + 8);
      v2f b3 = *(const v2f*)(bsh + ka + 12);
      c0 = wmma_f32_k4(a0, b0, c0);
      c1 = wmma_f32_k4(a1, b1, c1);
      c2 = wmma_f32_k4(a2, b2, c2);
      c3 = wmma_f32_k4(a3, b3, c3);
    }
    const v8f acc = (c0 + c1) + (c2 + c3);

#pragma unroll
    for (int i = 0; i < 8; ++i) {
      const int m = rowBase + i;
      const size_t oidx = ((size_t)m * kS + (size_t)s) * kH + nCol;
      const float v = tanhf(acc[i] + out[oidx] + bias);
      out[oidx] = v;
      hout[(size_t)m * kH + nCol] = v;
    }

    // ---- device-wide barrier: monotonic counter, release/acquire fences ----
    __syncthreads();
    __builtin_amdgcn_fence(__ATOMIC_RELEASE, "agent");
    if (threadIdx.x == 0) {
      __hip_atomic_fetch_add(bar, 1u, __ATOMIC_RELEASE, __HIP_MEMORY_SCOPE_AGENT);
      const unsigned tgt = (unsigned)kScanBlocks * (unsigned)(s + 1);
      while (__hip_atomic_load(bar, __ATOMIC_ACQUIRE,
                               __HIP_MEMORY_SCOPE_AGENT) < tgt) {
        __builtin_amdgcn_s_sleep(2);
      }
    }
    __syncthreads();
    __builtin_amdgcn_fence(__ATOMIC_ACQUIRE, "agent");
  }
}

extern "C" void kernel_launch(void* const* d_in, const int* in_sizes, int n_in,
                              void* d_out, int out_size, void* d_ws, size_t ws_size,
                              hipStream_t stream) {
  const float* x   = (const float*)d_in[0];  // [B,S,I]
  const float* h0  = (const float*)d_in[1];  // [B,H]
  const float* Wih = (const float*)d_in[2];  // [H,I]
  const float* bih = (const float*)d_in[3];  // [H]
  const float* Whh = (const float*)d_in[4];  // [H,H]
  const float* bhh = (const float*)d_in[5];  // [H]

  float* out = (float*)d_out;  // [B,S,H] output, then [B,H] h_final
  float* hb0 = (float*)d_ws;
  float* hb1 = hb0 + (size_t)kB * kH;
  unsigned* bar = (unsigned*)(hb1 + (size_t)kB * kH);

  // 1) Input projection (bulk GEMM): 2048 waves, 8 per block.
  rnn_xproj_wmma<<<kM1 / 16 * 16 / 128, 256, 0, stream>>>(x, Wih, bih, out);

  // 2) Persistent scan over all 512 timesteps (barrier counter zeroed first).
  (void)hipMemsetAsync(bar, 0, 64, stream);
  rnn_scan_persistent<<<kScanBlocks, 128, kLdsFloats * sizeof(float), stream>>>(
      h0, Whh, bhh, out, hb0, hb1, bar);

  // 3) h_final (step 511 wrote hb1) -> tail of d_out.
  (void)hipMemcpyAsync(out + (size_t)kB * kS * kH, hb1,
                       (size_t)kB * kH * sizeof(float),
                       hipMemcpyDeviceToDevice, stream);
}